// LocalAffineWarp_17763984736735
// MI455X (gfx1250) — compile-verified
//
#include <hip/hip_runtime.h>
#include <hip/hip_bf16.h>
#include <math.h>

// Problem constants (match reference)
#define BB   2
#define NN   5
#define CC   256
#define HH   64
#define WW   64
#define PARA 256
#define H0   60
#define W0   60
#define PI_F 3.14159265358979323846f

typedef float v2f __attribute__((ext_vector_type(2)));
typedef float v8f __attribute__((ext_vector_type(8)));

// ---------------------------------------------------------------------------
// One wave computes a 16(M) x 16(N) fp32 tile of  X(2xK) @ W(ROWS x K)^T
// using V_WMMA_F32_16X16X4_F32, K consumed 4 at a time.
// A layout (16x4 f32): lane L -> M = L&15 ; VGPR0/1 = K = k0 + (L>=16?2:0) (+1)
// B layout (4x16 f32): lane L -> N = L&15 ; VGPR0/1 = K = k0 + (L>=16?2:0) (+1)
// D layout: acc[r], lane L -> M = r + (L>=16?8:0), N = L&15
// Branchless: invalid M-rows (m >= 2) read row 0 and are zeroed by multiply,
// so EXEC stays all-ones through the whole loop (WMMA requirement) and no
// exec-mask save/restore branches are generated around the loads.
// ---------------------------------------------------------------------------
__device__ inline v8f wmma_rows2_tile(const float* __restrict__ X,
                                      const float* __restrict__ Wm,
                                      int n_base, int K)
{
    const int   lane = threadIdx.x & 31;
    const int   m    = lane & 15;
    const int   koff = (lane >> 4) << 1;          // 0 or 2
    const int   mrow = (m < BB) ? m : 0;          // safe row
    const float msel = (m < BB) ? 1.0f : 0.0f;    // zero pad rows

    const float* __restrict__ Xp = X  + mrow * K + koff;           // 8B aligned
    const float* __restrict__ Wp = Wm + (n_base + m) * K + koff;   // 8B aligned

    v8f acc = {};
    #pragma unroll 4
    for (int k0 = 0; k0 < K; k0 += 4) {
        v2f a = *(const v2f*)(Xp + k0);           // global_load_b64
        v2f b = *(const v2f*)(Wp + k0);           // global_load_b64
        a.x *= msel;
        a.y *= msel;
        acc = __builtin_amdgcn_wmma_f32_16x16x4_f32(
                  false, a, false, b, (short)0, acc, false, false);
    }
    return acc;
}

// K1: h = relu(para_code @ W1^T + b1)   -> h (2, 256)
__global__ __launch_bounds__(32)
void mlp_hidden_kernel(const float* __restrict__ pc,
                       const float* __restrict__ W1,
                       const float* __restrict__ b1,
                       float* __restrict__ h)
{
    const int n_base = blockIdx.x * 16;
    v8f acc = wmma_rows2_tile(pc, W1, n_base, PARA);
    const int lane = threadIdx.x & 31;
    if (lane < 16) {                         // rows M=0..7 live here; need 0,1
        const int n = n_base + lane;
        const float bia = b1[n];
        h[0 * PARA + n] = fmaxf(acc[0] + bia, 0.0f);
        h[1 * PARA + n] = fmaxf(acc[1] + bia, 0.0f);
    }
}

// K2: three heads.  blocks [0,160): Wk->kpts tanh ; [160,480): Ws->sim tanh ;
//                   [480,560): Wr->rad sigmoid.   Outputs stored flat (B, ROWS).
__global__ __launch_bounds__(32)
void mlp_heads_kernel(const float* __restrict__ h,
                      const float* __restrict__ Wk, const float* __restrict__ bk,
                      const float* __restrict__ Ws, const float* __restrict__ bs,
                      const float* __restrict__ Wr, const float* __restrict__ br,
                      float* __restrict__ kpts,   // (B, 2*N*C)
                      float* __restrict__ sim,    // (B, 4*N*C)
                      float* __restrict__ rad)    // (B, N*C)
{
    const int blk = blockIdx.x;
    const float* Wm; const float* bias; float* dst; int rows, n_base, act;
    if (blk < 160)      { Wm = Wk; bias = bk; dst = kpts; rows = 2*NN*CC; n_base = blk * 16;        act = 0; }
    else if (blk < 480) { Wm = Ws; bias = bs; dst = sim;  rows = 4*NN*CC; n_base = (blk-160) * 16;  act = 0; }
    else                { Wm = Wr; bias = br; dst = rad;  rows =   NN*CC; n_base = (blk-480) * 16;  act = 1; }

    v8f acc = wmma_rows2_tile(h, Wm, n_base, PARA);
    const int lane = threadIdx.x & 31;
    if (lane < 16) {
        const int n = n_base + lane;
        const float bia = bias[n];
        #pragma unroll
        for (int b = 0; b < BB; ++b) {
            float v = acc[b] + bia;
            v = act ? (1.0f / (1.0f + __expf(-v))) : tanhf(v);
            dst[b * rows + n] = v;
        }
    }
}

// K3: warp field + softmax-over-N reduction -> g2 (B, C, H0, W0, 2)
__global__ __launch_bounds__(256)
void warp_field_kernel(const float* __restrict__ kpts,  // (B, N*C*2)
                       const float* __restrict__ rad,   // (B, N*C)
                       const float* __restrict__ sim,   // (B, N*C*4)
                       const float* __restrict__ grid0, // (H0, W0, 2)
                       float* __restrict__ g2)          // (B, C, H0, W0, 2)
{
    const int t = blockIdx.x * blockDim.x + threadIdx.x;
    if (t >= BB * CC * H0 * W0) return;
    int r = t;
    const int j = r % W0; r /= W0;
    const int i = r % H0; r /= H0;
    const int c = r % CC; const int b = r / CC;

    const v2f g = *(const v2f*)(grid0 + (i * W0 + j) * 2);
    const float gx = g.x;
    const float gy = g.y;

    float wsum = 0.0f, ax = 0.0f, ay = 0.0f;
    #pragma unroll
    for (int n = 0; n < NN; ++n) {
        const int base = (b * NN + n) * CC + c;
        const v2f kp = *(const v2f*)(kpts + base * 2);
        const float kx = kp.x;
        const float ky = kp.y;
        const float rd = rad[base];
        const float s0 = sim[base * 4 + 0];
        const float s1 = sim[base * 4 + 1];
        const float s2 = sim[base * 4 + 2];
        const float s3 = sim[base * 4 + 3];

        const float ox = gx - kx, oy = gy - ky;
        const float dist = sqrtf(ox * ox + oy * oy);
        const float e = __expf(-dist / rd);
        const float scale = (s0 + 1.0f) * e + (1.0f - e);
        const float ang = s1 * PI_F * e;
        const float cs = __cosf(ang);
        const float sn = __sinf(ang);
        const float rx =  ox * cs + oy * sn;
        const float ry = -ox * sn + oy * cs;
        const float wx = scale * rx + s2 * e + kx;
        const float wy = scale * ry + s3 * e + ky;

        const float se = __expf(e);          // softmax over N applied to e
        wsum += se;
        ax += wx * se;
        ay += wy * se;
    }
    const float inv = 1.0f / wsum;
    v2f o; o.x = ax * inv; o.y = ay * inv;
    *(v2f*)(g2 + t * 2) = o;                 // global_store_b64
}

// Scrambled read of g2 implementing swapaxes/transpose/reshape:
// resized-src channel (c,jch) at (p,q)  <-  g2[b,c,h_src,w_src,comp]
__device__ inline float g2_scrambled(const float* __restrict__ g2,
                                     int bc, int jch, int p, int q)
{
    const int flat  = jch * (H0 * W0) + p * W0 + q;
    const int w_src = flat / (H0 * 2);
    const int rem   = flat - w_src * (H0 * 2);
    const int h_src = rem >> 1;
    const int comp  = rem & 1;
    return g2[(bc * (H0 * W0) + h_src * W0 + w_src) * 2 + comp];
}

// K4: fused (scramble + 60->64 bilinear resize) grid build + 3D trilinear sample
__global__ __launch_bounds__(256)
void resize_sample_kernel(const float* __restrict__ fmap, // (B, C, H, W)
                          const float* __restrict__ g2,   // (B, C, H0, W0, 2)
                          float* __restrict__ out)        // (B, C, H, W)
{
    const int t = blockIdx.x * blockDim.x + threadIdx.x;
    if (t >= BB * CC * HH * WW) return;
    int r = t;
    const int w = r % WW; r /= WW;
    const int h = r % HH; r /= HH;
    const int c = r % CC; const int b = r / CC;
    const int bc = b * CC + c;

    // bilinear 60->64 (half-pixel centers, edge clamp == jax.image.resize)
    const float scl = (float)H0 / (float)HH;     // 60/64, square
    const float sy = (h + 0.5f) * scl - 0.5f;
    const float sx = (w + 0.5f) * scl - 0.5f;
    const float syf = floorf(sy), sxf = floorf(sx);
    const float ty = sy - syf,   tx = sx - sxf;
    const int y0 = max((int)syf, 0),          x0 = max((int)sxf, 0);
    const int y1 = min((int)syf + 1, H0 - 1), x1 = min((int)sxf + 1, W0 - 1);

    float gv[2];
    #pragma unroll
    for (int j = 0; j < 2; ++j) {
        const float v00 = g2_scrambled(g2, bc, j, y0, x0);
        const float v01 = g2_scrambled(g2, bc, j, y0, x1);
        const float v10 = g2_scrambled(g2, bc, j, y1, x0);
        const float v11 = g2_scrambled(g2, bc, j, y1, x1);
        gv[j] = (1.0f - ty) * ((1.0f - tx) * v00 + tx * v01)
              +         ty  * ((1.0f - tx) * v10 + tx * v11);
    }

    // depth channel: linspace(-1+1/C, 1-1/C, C)
    const float z = -1.0f + (2.0f * (float)c + 1.0f) / (float)CC;

    // trilinear, zeros padding, align_corners=False
    const float xf = (gv[0] + 1.0f) * (WW * 0.5f) - 0.5f;
    const float yf = (gv[1] + 1.0f) * (HH * 0.5f) - 0.5f;
    const float zf = (z     + 1.0f) * (CC * 0.5f) - 0.5f;
    const float x0f = floorf(xf), y0f = floorf(yf), z0f = floorf(zf);
    const float fx = xf - x0f, fy = yf - y0f, fz = zf - z0f;
    const int xi0 = (int)x0f, yi0 = (int)y0f, zi0 = (int)z0f;

    // Branchless 8-tap: clamp indices for the address, zero the weight when
    // out of bounds. Loads are unconditional -> clean vmem clauses, no EXEC
    // divergence.
    float acc = 0.0f;
    #pragma unroll
    for (int dz = 0; dz < 2; ++dz) {
        const int zi = zi0 + dz;
        const int zic = min(max(zi, 0), CC - 1);
        const float wz = (dz ? fz : (1.0f - fz)) *
                         ((zi == zic) ? 1.0f : 0.0f);
        #pragma unroll
        for (int dy = 0; dy < 2; ++dy) {
            const int yi = yi0 + dy;
            const int yic = min(max(yi, 0), HH - 1);
            const float wy = (dy ? fy : (1.0f - fy)) *
                             ((yi == yic) ? 1.0f : 0.0f);
            #pragma unroll
            for (int dx = 0; dx < 2; ++dx) {
                const int xi = xi0 + dx;
                const int xic = min(max(xi, 0), WW - 1);
                const float wx = (dx ? fx : (1.0f - fx)) *
                                 ((xi == xic) ? 1.0f : 0.0f);
                acc += wz * wy * wx *
                       fmap[((b * CC + zic) * HH + yic) * WW + xic];
            }
        }
    }
    out[t] = acc;
}

// ---------------------------------------------------------------------------
extern "C" void kernel_launch(void* const* d_in, const int* in_sizes, int n_in,
                              void* d_out, int out_size, void* d_ws, size_t ws_size,
                              hipStream_t stream)
{
    const float* fmap  = (const float*)d_in[0];   // (B,C,H,W)
    const float* pc    = (const float*)d_in[1];   // (B,PARA)
    const float* W1    = (const float*)d_in[2];   // (PARA,PARA)
    const float* b1    = (const float*)d_in[3];
    const float* Wk    = (const float*)d_in[4];   // (2NC,PARA)
    const float* bk    = (const float*)d_in[5];
    const float* Ws    = (const float*)d_in[6];   // (4NC,PARA)
    const float* bs    = (const float*)d_in[7];
    const float* Wr    = (const float*)d_in[8];   // (NC,PARA)
    const float* br    = (const float*)d_in[9];
    const float* grid0 = (const float*)d_in[10];  // (H0,W0,2)
    float* out = (float*)d_out;

    float* ws = (float*)d_ws;
    float* h    = ws;                    // 2*256            = 512
    float* kpts = h    + BB * PARA;      // 2*2560           = 5120
    float* rad  = kpts + BB * 2*NN*CC;   // 2*1280           = 2560
    float* sim  = rad  + BB * NN*CC;     // 2*5120           = 10240
    float* g2   = sim  + BB * 4*NN*CC;   // 2*256*60*60*2    = 3,686,400
    (void)ws_size; (void)in_sizes; (void)n_in; (void)out_size;

    // K1: hidden layer (WMMA fp32)
    mlp_hidden_kernel<<<PARA / 16, 32, 0, stream>>>(pc, W1, b1, h);

    // K2: heads (WMMA fp32): 160 + 320 + 80 tiles
    mlp_heads_kernel<<<560, 32, 0, stream>>>(h, Wk, bk, Ws, bs, Wr, br,
                                             kpts, sim, rad);

    // K3: warp field + softmax reduction
    {
        const int n = BB * CC * H0 * W0;
        warp_field_kernel<<<(n + 255) / 256, 256, 0, stream>>>(kpts, rad, sim,
                                                               grid0, g2);
    }

    // K4: fused scramble + bilinear resize + trilinear grid-sample
    {
        const int n = BB * CC * HH * WW;
        resize_sample_kernel<<<(n + 255) / 256, 256, 0, stream>>>(fmap, g2, out);
    }
}